// DLI_loss_1_6614249636376
// MI455X (gfx1250) — compile-verified
//
#include <hip/hip_runtime.h>
#include <hip/hip_bf16.h>
#include <math.h>

#define BSZ  32
#define SEQ  4096
#define ENC  512
#define TMAX 64

typedef __attribute__((ext_vector_type(2))) float v2f;
typedef __attribute__((ext_vector_type(8))) float v8f;

// ---------------------------------------------------------------------------
// Kernel 1: fused gather + projection using V_WMMA_F32_16X16X4_F32.
// Grid: BSZ*4 blocks of 32 threads (1 wave each). Wave w of batch b computes
// rows t0 = w*16 .. t0+15 of proj[b][t][4] where
//   proj[..][0..1] = left  = his . Wl^T   (Wl = W[:, :ENC])
//   proj[..][2..3] = right = his . Wr^T   (Wr = W[:, ENC:])
// A fragment (16x4 f32, ISA layout): lanes 0-15 hold M=lane, K=koff{0},koff+1;
// lanes 16-31 hold M=lane-16, K=2,3. B fragment mirrors for 4x16 (N across
// lanes, K across the 2 VGPRs / lane-halves). N is padded 4 -> 16 with zeros.
// ---------------------------------------------------------------------------
__global__ __launch_bounds__(32)
void proj_wmma_kernel(const float* __restrict__ enc_out,
                      const int*   __restrict__ end_ids,
                      const float* __restrict__ W,
                      float*       __restrict__ proj)
{
    const int b    = blockIdx.x >> 2;
    const int wave = blockIdx.x & 3;
    const int lane = threadIdx.x & 31;

    const int t0   = wave * 16;
    const int mrow = lane & 15;            // A-row within the tile
    const int koff = (lane >> 4) * 2;      // 0 (lanes 0-15) or 2 (lanes 16-31)

    // Gathered sequence row for this lane's A rows
    const int sidx = end_ids[b * TMAX + (t0 + mrow)];
    const float* arow = enc_out + ((size_t)b * SEQ + (size_t)sidx) * ENC;

    // B operand column n: 0 -> W[0][k], 1 -> W[1][k], 2 -> W[0][ENC+k],
    // 3 -> W[1][ENC+k], n>=4 -> 0. Clamp address so the load is always legal;
    // zero via select (no branch -> EXEC stays all-ones for WMMA).
    const int  n       = lane & 15;
    const int  nc      = (n < 4) ? n : 3;
    const bool bvalid  = (n < 4);
    const float* wptr  = W + (size_t)(nc & 1) * (2 * ENC) + (size_t)(nc >> 1) * ENC;

    v8f acc = {};
#if __has_builtin(__builtin_amdgcn_wmma_f32_16x16x4_f32)
    for (int k = 0; k < ENC; k += 4) {
        v2f a;
        a.x = arow[k + koff];
        a.y = arow[k + koff + 1];
        float w0 = wptr[k + koff];
        float w1 = wptr[k + koff + 1];
        v2f bb;
        bb.x = bvalid ? w0 : 0.0f;
        bb.y = bvalid ? w1 : 0.0f;
        acc = __builtin_amdgcn_wmma_f32_16x16x4_f32(
            /*neg_a=*/false, a, /*neg_b=*/false, bb,
            /*c_mod=*/(short)0, acc, /*reuse_a=*/false, /*reuse_b=*/false);
    }
#else
    // Scalar fallback (shows wmma=0 in histogram -> signal to switch path)
    for (int k = 0; k < ENC; k += 4) {
        float a0 = arow[k + koff];
        float a1 = arow[k + koff + 1];
        float w0 = bvalid ? wptr[k + koff]     : 0.0f;
        float w1 = bvalid ? wptr[k + koff + 1] : 0.0f;
        acc[0] += a0 * w0 + a1 * w1;   // placeholder, not layout-exact
    }
#endif

    // D layout (16x16 f32): VGPR r, lanes 0-15 -> M=r, lanes 16-31 -> M=r+8,
    // N = lane&15. Only N < 4 columns are real outputs.
    if (n < 4) {
        const int mbase = (lane >> 4) * 8;
#pragma unroll
        for (int r = 0; r < 8; ++r) {
            const int m = mbase + r;
            proj[((size_t)b * TMAX + (t0 + m)) * 4 + n] = acc[r];
        }
    }
}

// ---------------------------------------------------------------------------
// Kernel 2: per-batch masked CE partial sums. One block per batch, thread = j.
// logits[b,j,k,o] = left[b,j,o] + right[b,k,o] + bias[o]; 2-class log-softmax.
// valid = (k<j) && (j < turn_len[b]); label = (k == j-1).
// ---------------------------------------------------------------------------
__global__ __launch_bounds__(TMAX)
void loss_partial_kernel(const float* __restrict__ proj,
                         const int*   __restrict__ turn_len,
                         const float* __restrict__ bias,
                         float*       __restrict__ partial_sum,
                         int*         __restrict__ partial_cnt)
{
    const int b = blockIdx.x;
    const int j = threadIdx.x;

    const int   L  = turn_len[b];
    const float b0 = bias[0];
    const float b1 = bias[1];

    const float l0 = proj[((size_t)b * TMAX + j) * 4 + 0] + b0;
    const float l1 = proj[((size_t)b * TMAX + j) * 4 + 1] + b1;

    float s = 0.0f;
    int   c = 0;
    const bool jvalid = (j < L);

    for (int k = 0; k < TMAX; ++k) {
        if (jvalid && (k < j)) {
            const float r0 = proj[((size_t)b * TMAX + k) * 4 + 2];
            const float r1 = proj[((size_t)b * TMAX + k) * 4 + 3];
            const float z0 = l0 + r0;
            const float z1 = l1 + r1;
            // stable 2-class logsumexp
            const float m   = fmaxf(z0, z1);
            const float lse = m + log1pf(expf(-fabsf(z0 - z1)));
            const float lp  = ((k == j - 1) ? z1 : z0) - lse;
            s += -lp;
            c += 1;
        }
    }

    __shared__ float ssum[TMAX];
    __shared__ int   scnt[TMAX];
    ssum[j] = s;
    scnt[j] = c;
    __syncthreads();
    for (int off = TMAX / 2; off > 0; off >>= 1) {
        if (j < off) {
            ssum[j] += ssum[j + off];
            scnt[j] += scnt[j + off];
        }
        __syncthreads();
    }
    if (j == 0) {
        partial_sum[b] = ssum[0];
        partial_cnt[b] = scnt[0];
    }
}

// ---------------------------------------------------------------------------
// Kernel 3: deterministic serial reduction of 32 partials -> scalar loss.
// ---------------------------------------------------------------------------
__global__ void finalize_kernel(const float* __restrict__ partial_sum,
                                const int*   __restrict__ partial_cnt,
                                float*       __restrict__ out)
{
    float s = 0.0f;
    int   c = 0;
    for (int b = 0; b < BSZ; ++b) {
        s += partial_sum[b];
        c += partial_cnt[b];
    }
    out[0] = s / (float)(c < 1 ? 1 : c);
}

extern "C" void kernel_launch(void* const* d_in, const int* in_sizes, int n_in,
                              void* d_out, int out_size, void* d_ws, size_t ws_size,
                              hipStream_t stream)
{
    const float* enc_out  = (const float*)d_in[0];   // (32, 4096, 512) f32
    const int*   end_ids  = (const int*)  d_in[1];   // (32, 64) i32
    const int*   turn_len = (const int*)  d_in[2];   // (32,) i32
    const float* W        = (const float*)d_in[3];   // (2, 1024) f32
    const float* bias     = (const float*)d_in[4];   // (2,) f32
    float*       out      = (float*)d_out;           // scalar f32

    // Workspace layout:
    //   [0,   128)  partial_sum : 32 floats
    //   [128, 256)  partial_cnt : 32 ints
    //   [256, ...)  proj        : 32*64*4 floats = 32 KB
    char*  ws          = (char*)d_ws;
    float* partial_sum = (float*)(ws);
    int*   partial_cnt = (int*)  (ws + 128);
    float* proj        = (float*)(ws + 256);

    proj_wmma_kernel<<<dim3(BSZ * 4), dim3(32), 0, stream>>>(enc_out, end_ids, W, proj);
    loss_partial_kernel<<<dim3(BSZ), dim3(TMAX), 0, stream>>>(proj, turn_len, bias,
                                                              partial_sum, partial_cnt);
    finalize_kernel<<<dim3(1), dim3(1), 0, stream>>>(partial_sum, partial_cnt, out);
}